// MDSPreimage_8959301779821
// MI455X (gfx1250) — compile-verified
//
#include <hip/hip_runtime.h>

// MDS preimage: b=4096 rows, n=50000 candidates, c=64 channels, topk=128.
// One workgroup (512 threads = 16 wave32) per row:
//   1) stage the whole 200KB distance row into LDS
//        - preferred: single TDM descriptor (tensor_load_to_lds, TENSORcnt)
//        - fallback : vectorized global->VGPR->LDS copy
//   2) 4-pass MSB radix-select on monotone keys (computed on the fly)
//   3) collect (key,idx), deterministic rank-sort to match top_k ordering
//   4) gather y rows into LDS
//        - preferred: global_load_async_to_lds_b128 (ASYNCcnt)
//        - fallback : float4 loads + LDS stores
//   5) y_outer = yc^T yc via v_wmma_f32_16x16x4_f32 (16 tiles, 1 per wave)
//   6) 64x64 Cholesky + tri-solves in LDS, write preimage + indices

#define KSEL     128
#define CDIM     64
#define YSTR     68    // padded LDS stride for yc tile (bank-conflict free)
#define CSTR     65    // padded stride for y_outer
#define NTHREADS 512

typedef float    v2f __attribute__((ext_vector_type(2)));
typedef float    v8f __attribute__((ext_vector_type(8)));
typedef unsigned v4u __attribute__((ext_vector_type(4)));
typedef int      v4i __attribute__((ext_vector_type(4)));
typedef int      v8i __attribute__((ext_vector_type(8)));

#if defined(__has_builtin)
#if __has_builtin(__builtin_amdgcn_tensor_load_to_lds) && \
    __has_builtin(__builtin_amdgcn_s_wait_tensorcnt)
#define USE_TDM 1
#endif
#if __has_builtin(__builtin_amdgcn_global_load_async_to_lds_b128) && \
    __has_builtin(__builtin_amdgcn_s_wait_asynccnt)
#define USE_ASYNC 1
#endif
#endif

// builtin pointer types: v4i in global (AS1) / LDS (AS3) address spaces
typedef __attribute__((address_space(1))) v4i* gv4i_p;
typedef __attribute__((address_space(3))) v4i* lv4i_p;

// monotone order-preserving transform on raw float bits
__device__ __forceinline__ unsigned bits2key(unsigned u) {
    return (u & 0x80000000u) ? ~u : (u | 0x80000000u);
}
__device__ __forceinline__ float key2f(unsigned k) {
    unsigned u = (k & 0x80000000u) ? (k & 0x7FFFFFFFu) : ~k;
    return __uint_as_float(u);
}

__global__ __launch_bounds__(NTHREADS)
void mds_preimage_kernel(const float* __restrict__ sq_dist,
                         const float* __restrict__ y,
                         float* __restrict__ out_pre,
                         int*   __restrict__ out_idx,
                         int n)
{
    extern __shared__ unsigned char smem_dyn[];
    unsigned* keys = (unsigned*)smem_dyn;          // phase 1-3: n raw float bits
    float*    yc   = (float*)smem_dyn;             // phase 4+ : KSEL x YSTR tile

    __shared__ unsigned hist[256];
    __shared__ unsigned sel_key[KSEL];
    __shared__ int      sel_idx[KSEL];
    __shared__ int      sidx[KSEL];
    __shared__ float    w[KSEL];
    __shared__ float    diag[KSEL];
    __shared__ float    mean[CDIM];
    __shared__ float    rhs[CDIM];
    __shared__ float    zvec[CDIM];
    __shared__ float    xvec[CDIM];
    __shared__ float    youter[CDIM * CSTR];
    __shared__ unsigned sh_prefix;
    __shared__ int      sh_remain;
    __shared__ int      cnt_less, cnt_eq;
    __shared__ float    sh_pivot;

    const int tid = threadIdx.x;
    const int row = blockIdx.x;
    const float* drow = sq_dist + (size_t)row * (size_t)n;

    // ---------------- 1) stage raw row bits into LDS ----------------
    bool staged = false;
#if defined(USE_TDM)
    if (n <= 65535) {                    // fits TDM 16-bit tile_dim0
        if (tid == 0) {
            unsigned lds_base = (unsigned)(size_t)(void*)keys;
            unsigned long long ga = (unsigned long long)(size_t)drow;
            v4u g0;
            g0[0] = 1u;                              // count=1, user descriptor
            g0[1] = lds_base;                        // lds_addr
            g0[2] = (unsigned)ga;                    // global_addr[31:0]
            g0[3] = (unsigned)((ga >> 32) & 0x1FFFFFFu) | 0x80000000u; // addr[56:32] | type=2
            v8i g1;
            g1[0] = 0x00020000;                      // data_size=4B, no flags
            g1[1] = (int)(((unsigned)n & 0xFFFFu) << 16);        // tensor_dim0 lo
            g1[2] = (int)(((unsigned)n >> 16) | (1u << 16));     // tensor_dim0 hi, tensor_dim1=1
            g1[3] = (int)(((unsigned)n & 0xFFFFu) << 16);        // tile_dim0 = n
            g1[4] = 1;                               // tile_dim1 = 1
            g1[5] = n;                               // tensor_dim0_stride lo
            g1[6] = 0;
            g1[7] = 0;
            v4i gz4 = {0, 0, 0, 0};
            v8i gz8 = {0, 0, 0, 0, 0, 0, 0, 0};
            __builtin_amdgcn_tensor_load_to_lds(g0, g1, gz4, gz4, gz8, 0);
            __builtin_amdgcn_s_wait_tensorcnt(0);
        }
        staged = true;
    }
#endif
    if (!staged) {
        if ((n & 3) == 0) {
            const uint4* drow4 = (const uint4*)drow;
            uint4* keys4 = (uint4*)keys;
            int n4 = n >> 2;
            for (int i = tid; i < n4; i += NTHREADS) keys4[i] = drow4[i];
        } else {
            const unsigned* drowu = (const unsigned*)drow;
            for (int i = tid; i < n; i += NTHREADS) keys[i] = drowu[i];
        }
    }
    __syncthreads();

    // ---------------- 2) MSB radix select (4 x 8-bit passes) ----------------
    unsigned prefix = 0;
    int remaining = KSEL;
    for (int shift = 24; shift >= 0; shift -= 8) {
        if (tid < 256) hist[tid] = 0u;
        __syncthreads();
        unsigned himask = (shift == 24) ? 0u : (0xFFFFFFFFu << (shift + 8));
        for (int i = tid; i < n; i += NTHREADS) {
            unsigned kk = bits2key(keys[i]);
            if ((kk & himask) == (prefix & himask))
                atomicAdd(&hist[(kk >> shift) & 255u], 1u);
        }
        __syncthreads();
        if (tid == 0) {
            unsigned cum = 0; int bsel = 0;
            for (; bsel < 256; ++bsel) {
                unsigned h = hist[bsel];
                if (cum + h >= (unsigned)remaining) break;
                cum += h;
            }
            sh_prefix = prefix | ((unsigned)bsel << shift);
            sh_remain = remaining - (int)cum;
        }
        __syncthreads();
        prefix = sh_prefix;
        remaining = sh_remain;
    }
    const unsigned T = prefix;          // exact 128th-smallest key
    // remaining = how many elements equal to T we still need

    // ---------------- 3) collect + rank sort ----------------
    if (tid == 0) { cnt_less = 0; cnt_eq = 0; }
    __syncthreads();
    {
        const int base_eq = KSEL - remaining;
        for (int i = tid; i < n; i += NTHREADS) {
            unsigned kk = bits2key(keys[i]);
            if (kk < T) {
                int p = atomicAdd(&cnt_less, 1);
                sel_key[p] = kk; sel_idx[p] = i;
            } else if (kk == T) {
                int p = atomicAdd(&cnt_eq, 1);
                if (p < remaining) { sel_key[base_eq + p] = kk; sel_idx[base_eq + p] = i; }
            }
        }
    }
    __syncthreads();
    if (tid < KSEL) {
        unsigned mk = sel_key[tid]; int mi = sel_idx[tid];
        int r = 0;
        for (int j = 0; j < KSEL; ++j) {
            unsigned kj = sel_key[j]; int ij = sel_idx[j];
            if (kj < mk || (kj == mk && ij < mi)) ++r;
        }
        sidx[r] = mi;
        w[r]    = key2f(mk);
    }
    __syncthreads();   // selection finished; safe to overwrite `keys` region

    // ---------------- 4) gather y rows, center, diag, rhs ----------------
#if defined(USE_ASYNC)
    for (int e = tid; e < KSEL * (CDIM / 4); e += NTHREADS) {
        int r  = e >> 4;
        int c4 = e & 15;
        float* src = (float*)(y + (size_t)sidx[r] * CDIM + c4 * 4);
        float* dst = &yc[r * YSTR + c4 * 4];
        __builtin_amdgcn_global_load_async_to_lds_b128((gv4i_p)src, (lv4i_p)dst, 0, 0);
    }
    __builtin_amdgcn_s_wait_asynccnt(0);
#else
    {
        const float4* y4 = (const float4*)y;      // CDIM==64 -> 16 float4/row
        for (int e = tid; e < KSEL * (CDIM / 4); e += NTHREADS) {
            int r  = e >> 4;
            int c4 = e & 15;
            float4 v = y4[(size_t)sidx[r] * (CDIM / 4) + c4];
            *(float4*)&yc[r * YSTR + c4 * 4] = v;
        }
    }
#endif
    __syncthreads();
    if (tid < CDIM) {
        float s = 0.f;
        for (int r = 0; r < KSEL; ++r) s += yc[r * YSTR + tid];
        mean[tid] = s * (1.0f / (float)KSEL);
    }
    __syncthreads();
    for (int e = tid; e < KSEL * CDIM; e += NTHREADS) {
        int r = e >> 6, c = e & 63;
        yc[r * YSTR + c] -= mean[c];
    }
    __syncthreads();
    if (tid < KSEL) {
        float s = 0.f;
        for (int c = 0; c < CDIM; ++c) { float v = yc[tid * YSTR + c]; s += v * v; }
        diag[tid] = s;
    }
    __syncthreads();
    if (tid < CDIM) {
        float s = 0.f;
        for (int r = 0; r < KSEL; ++r) s += yc[r * YSTR + tid] * (diag[r] - w[r]);
        rhs[tid] = s;
    }
    __syncthreads();

    // ---------------- 5) y_outer = yc^T yc via WMMA fp32 ----------------
    {
        const int wave = tid >> 5;
        const int lane = tid & 31;
        const int ti = wave >> 2;       // C row-tile (0..3)
        const int tj = wave & 3;        // C col-tile (0..3)
        const int m    = lane & 15;
        const int koff = (lane >> 4) << 1;   // 16x4 fp32 A layout: K 0/1 vs 2/3
        const int ci = ti * 16 + m;
        const int cj = tj * 16 + m;

        v8f acc = {};
        for (int k0 = 0; k0 < KSEL; k0 += 4) {
            int ra = k0 + koff;
            v2f A, B;
            A.x = yc[ra * YSTR + ci];
            A.y = yc[(ra + 1) * YSTR + ci];
            B.x = yc[ra * YSTR + cj];
            B.y = yc[(ra + 1) * YSTR + cj];
            acc = __builtin_amdgcn_wmma_f32_16x16x4_f32(
                false, A, false, B, (short)0, acc, false, false);
        }
        const int rbase = ti * 16 + ((lane >> 4) << 3);
        const int col   = tj * 16 + (lane & 15);
#pragma unroll
        for (int r = 0; r < 8; ++r)
            youter[(rbase + r) * CSTR + col] = acc[r];
    }
    __syncthreads();

    // ---------------- 6) Cholesky + solves (lower triangle) ----------------
    for (int j = 0; j < CDIM; ++j) {
        __syncthreads();
        if (tid == 0) sh_pivot = sqrtf(youter[j * CSTR + j]);
        __syncthreads();
        float dj = sh_pivot;
        if (tid == j) youter[j * CSTR + j] = dj;
        if (tid > j && tid < CDIM) youter[tid * CSTR + j] /= dj;
        __syncthreads();
        if (tid > j && tid < CDIM) {
            float ljt = youter[tid * CSTR + j];
            for (int c2 = j + 1; c2 <= tid; ++c2)
                youter[tid * CSTR + c2] -= ljt * youter[c2 * CSTR + j];
        }
    }
    // forward: L z = rhs
    for (int j = 0; j < CDIM; ++j) {
        __syncthreads();
        if (tid == j) zvec[j] = rhs[j] / youter[j * CSTR + j];
        __syncthreads();
        if (tid > j && tid < CDIM) rhs[tid] -= youter[tid * CSTR + j] * zvec[j];
    }
    // backward: L^T x = z
    for (int j = CDIM - 1; j >= 0; --j) {
        __syncthreads();
        if (tid == j) xvec[j] = zvec[j] / youter[j * CSTR + j];
        __syncthreads();
        if (tid < j) zvec[tid] -= youter[j * CSTR + tid] * xvec[j];
    }
    __syncthreads();

    // ---------------- 7) outputs ----------------
    if (tid < CDIM)
        out_pre[(size_t)row * CDIM + tid] = 0.5f * xvec[tid] + mean[tid];
    if (tid < KSEL)
        out_idx[(size_t)row * KSEL + tid] = sidx[tid];
}

extern "C" void kernel_launch(void* const* d_in, const int* in_sizes, int n_in,
                              void* d_out, int out_size, void* d_ws, size_t ws_size,
                              hipStream_t stream) {
    const float* sq_dist = (const float*)d_in[0];
    const float* y       = (const float*)d_in[1];
    // reference shapes: sq_dist [b,n], y [n,64], topk == 128 (compile-time)
    const int n = in_sizes[1] / CDIM;
    const int b = in_sizes[0] / n;

    float* out_pre = (float*)d_out;                                   // [b,64] f32
    int*   out_idx = (int*)((float*)d_out + (size_t)b * CDIM);        // [b,128] i32

    size_t stage = (size_t)n * sizeof(unsigned);                      // topk staging
    size_t tile  = (size_t)KSEL * YSTR * sizeof(float);               // yc tile
    size_t shmem = (stage > tile ? stage : tile);
    shmem = (shmem + 15) & ~(size_t)15;

    mds_preimage_kernel<<<dim3(b), dim3(NTHREADS), shmem, stream>>>(
        sq_dist, y, out_pre, out_idx, n);
}